// ChebychevConvolution_79602923864565
// MI455X (gfx1250) — compile-verified
//
#include <hip/hip_runtime.h>

// ---------------------------------------------------------------------------
// Anisotropic Chebyshev convolution, fused as three WMMA GEMM phases.
//   a[b,k,i,m,q] = sum_p cheb1[k,p,m] * x[b,i,p,q]          (phase 1)
//   c[b,l,u,m,q] = sum_{k,i} coefs[k,l,i,u] * a[b,k,i,m,q]  (phase 2)
//   out[b,u,m,n] = sum_{l,q} c[b,l,u,m,q] * cheb2[l,q,n]    (phase 3)
// FP32 end-to-end via V_WMMA_F32_16X16X4_F32 (wave32 CDNA5 matrix op).
// ---------------------------------------------------------------------------

typedef float v2f __attribute__((ext_vector_type(2)));
typedef float v8f __attribute__((ext_vector_type(8)));

static constexpr int Bb = 8;
static constexpr int Cc = 32;
static constexpr int Uu = 32;
static constexpr int Kp = 5;      // Chebyshev powers
static constexpr int Nn = 256;    // N1 == N2
static constexpr int NN = Nn * Nn;        // 65536 spatial elements
static constexpr int KI = Kp * Cc;        // 160 = (k,i) fused dim
static constexpr int LU = Kp * Uu;        // 160 = (l,u) fused dim

__device__ __forceinline__ v8f wmma4(v2f a, v2f b, v8f c) {
  // D = A(16x4 f32) * B(4x16 f32) + C(16x16 f32)
  return __builtin_amdgcn_wmma_f32_16x16x4_f32(
      /*neg_a=*/false, a, /*neg_b=*/false, b,
      /*c_mod=*/(short)0, c, /*reuse_a=*/false, /*reuse_b=*/false);
}

// ---------------------------------------------------------------------------
// Phase 1: a[(b,k,i)] = cheb1[k]^T (256x256) @ x[b,i] (256x256)
// grid = (16 output tiles of 64x64, 1280 (b,k,i) combos), block = 256 (8 waves)
// ---------------------------------------------------------------------------
__global__ __launch_bounds__(256) void phase1_kernel(
    const float* __restrict__ cheb1, const float* __restrict__ x,
    float* __restrict__ a_out) {
  // K-innermost LDS tiles: As[m][p], Bs[q][p]; pad 18 keeps rows 8B aligned.
  __shared__ float As[64][18];
  __shared__ float Bs[64][18];

  const int combo = blockIdx.y;              // ((b*Kp)+k)*Cc + i
  const int i = combo % Cc;
  const int k = (combo / Cc) % Kp;
  const int b = combo / (Cc * Kp);
  const int m0 = (blockIdx.x & 3) * 64;
  const int q0 = (blockIdx.x >> 2) * 64;

  const float* chk = cheb1 + (size_t)k * NN;            // [p][m] row-major
  const float* xbi = x + ((size_t)(b * Cc + i)) * NN;   // [p][q] row-major

  const int t = threadIdx.x;
  const int lane = t & 31;
  const int w = t >> 5;
  const int tm = w & 3;                 // 4 m-subtiles
  const int tq = (w >> 2) << 1;         // waves 0-3 -> cols {0,1}, 4-7 -> {2,3}
  const int l15 = lane & 15;
  const int hi = lane >> 4;             // half-wave selects K pair
  const int row = tm * 16 + l15;
  const int c0 = tq * 16 + l15;

  const int pl = t >> 4;                // 0..15 : p within chunk
  const int e4 = (t & 15) * 4;          // 0..60 : m/q within tile

  v8f acc0 = {}; v8f acc1 = {};

  for (int pc = 0; pc < Nn; pc += 16) {
    // Stage A: As[m][p] = cheb1[k][pc+p][m0+m] (global contiguous along m)
    {
      float4 v = *(const float4*)&chk[(size_t)(pc + pl) * Nn + m0 + e4];
      As[e4 + 0][pl] = v.x; As[e4 + 1][pl] = v.y;
      As[e4 + 2][pl] = v.z; As[e4 + 3][pl] = v.w;
    }
    // Stage B: Bs[q][p] = x[b,i][pc+p][q0+q] (global contiguous along q)
    {
      float4 v = *(const float4*)&xbi[(size_t)(pc + pl) * Nn + q0 + e4];
      Bs[e4 + 0][pl] = v.x; Bs[e4 + 1][pl] = v.y;
      Bs[e4 + 2][pl] = v.z; Bs[e4 + 3][pl] = v.w;
    }
    __syncthreads();
#pragma unroll
    for (int kk = 0; kk < 4; ++kk) {
      const int p = kk * 4 + hi * 2;
      v2f av = *(const v2f*)&As[row][p];
      v2f b0 = *(const v2f*)&Bs[c0][p];
      v2f b1 = *(const v2f*)&Bs[c0 + 16][p];
      acc0 = wmma4(av, b0, acc0);
      acc1 = wmma4(av, b1, acc1);
    }
    __syncthreads();
  }

  float* op = a_out + (size_t)combo * NN;
#pragma unroll
  for (int r = 0; r < 8; ++r) {
    const int mg = m0 + tm * 16 + r + 8 * hi;
    op[(size_t)mg * Nn + q0 + c0] = acc0[r];
    op[(size_t)mg * Nn + q0 + c0 + 16] = acc1[r];
  }
}

// ---------------------------------------------------------------------------
// Phase 2: per batch b, c[(l,u), s] = sum_{(k,i)} coefs[k,l,i,u] * a[(k,i), s]
// M = 160 (l,u), K = 160 (k,i), N = 65536 spatial.
// grid = (512 s-tiles of 128, 5 lu-tiles of 32, 8 batches), block = 256.
// ---------------------------------------------------------------------------
__global__ __launch_bounds__(256) void phase2_kernel(
    const float* __restrict__ coefs, const float* __restrict__ a_in,
    float* __restrict__ c_out) {
  __shared__ float Wa[32][18];    // A[m'][k'] tile (k' innermost)
  __shared__ float Wb[128][18];   // B[s][k'] tile

  const int b = blockIdx.z;
  const int lblk = blockIdx.y;            // l index (m'-tile of 32 = one l)
  const int s0 = blockIdx.x * 128;

  const float* ab = a_in + (size_t)b * KI * NN;

  const int t = threadIdx.x;
  const int lane = t & 31;
  const int w = t >> 5;
  const int tm = w & 1;                   // 2 m'-subtiles
  const int ts = w >> 1;                  // 4 s-groups of 32
  const int l15 = lane & 15;
  const int hi = lane >> 4;
  const int row = tm * 16 + l15;
  const int c0 = ts * 32 + l15;

  v8f acc0 = {}; v8f acc1 = {};

  for (int kc = 0; kc < KI / 16; ++kc) {  // 10 chunks of 16 over (k,i)
    const int khi = kc >> 1;              // k (powers of graph 1)
    const int klo0 = (kc & 1) * 16;       // i offset within channel block
    // Stage A: Wa[m'][j] = coefs[khi, lblk, klo0+j, m'] (contiguous along u)
    if (t < 128) {
      const int j = t >> 3;               // 0..15
      const int m4 = (t & 7) * 4;         // 0..28
      float4 v = *(const float4*)&coefs[(size_t)khi * (Kp * Cc * Uu) +
                                        (size_t)lblk * (Cc * Uu) +
                                        (size_t)(klo0 + j) * Uu + m4];
      Wa[m4 + 0][j] = v.x; Wa[m4 + 1][j] = v.y;
      Wa[m4 + 2][j] = v.z; Wa[m4 + 3][j] = v.w;
    }
    // Stage B: Wb[s][j] = a[b][kc*16+j][s0+s] (contiguous along s)
    {
      const int j = t >> 4;               // 0..15
      const int s4 = (t & 15) * 8;        // 0..120, two float4 each
      const float* gp = ab + (size_t)(kc * 16 + j) * NN + s0 + s4;
      float4 v0 = *(const float4*)gp;
      float4 v1 = *(const float4*)(gp + 4);
      Wb[s4 + 0][j] = v0.x; Wb[s4 + 1][j] = v0.y;
      Wb[s4 + 2][j] = v0.z; Wb[s4 + 3][j] = v0.w;
      Wb[s4 + 4][j] = v1.x; Wb[s4 + 5][j] = v1.y;
      Wb[s4 + 6][j] = v1.z; Wb[s4 + 7][j] = v1.w;
    }
    __syncthreads();
#pragma unroll
    for (int kk = 0; kk < 4; ++kk) {
      const int p = kk * 4 + hi * 2;
      v2f av = *(const v2f*)&Wa[row][p];
      v2f b0 = *(const v2f*)&Wb[c0][p];
      v2f b1 = *(const v2f*)&Wb[c0 + 16][p];
      acc0 = wmma4(av, b0, acc0);
      acc1 = wmma4(av, b1, acc1);
    }
    __syncthreads();
  }

#pragma unroll
  for (int r = 0; r < 8; ++r) {
    const int mg = lblk * 32 + tm * 16 + r + 8 * hi;   // (l,u) fused index
    c_out[((size_t)b * LU + mg) * NN + s0 + c0] = acc0[r];
    c_out[((size_t)b * LU + mg) * NN + s0 + c0 + 16] = acc1[r];
  }
}

// ---------------------------------------------------------------------------
// Phase 3: per (b,u): out = sum_l c[b,l,u] (m x q) @ cheb2[l] (q x n)
// M = 256, N = 256, K = 5*256 = 1280.
// grid = (16 output tiles of 64x64, 256 (b,u) combos), block = 256.
// ---------------------------------------------------------------------------
__global__ __launch_bounds__(256) void phase3_kernel(
    const float* __restrict__ c_in, const float* __restrict__ cheb2,
    float* __restrict__ out) {
  __shared__ __align__(16) float Am[64][20];  // 80B rows -> b128 LDS stores
  __shared__ float Bn[64][18];

  const int bu = blockIdx.y;
  const int b = bu >> 5;
  const int u = bu & 31;
  const int m0 = (blockIdx.x & 3) * 64;
  const int n0 = (blockIdx.x >> 2) * 64;

  const int t = threadIdx.x;
  const int lane = t & 31;
  const int w = t >> 5;
  const int tm = w & 3;
  const int tn = (w >> 2) << 1;
  const int l15 = lane & 15;
  const int hi = lane >> 4;
  const int row = tm * 16 + l15;
  const int c0 = tn * 16 + l15;

  v8f acc0 = {}; v8f acc1 = {};

  for (int kc = 0; kc < (Kp * Nn) / 16; ++kc) {  // 80 chunks over (l,q)
    const int l = kc >> 4;                       // 256/16 chunks per l
    const int q0c = (kc & 15) * 16;
    const float* cb = c_in + ((size_t)((b * Kp + l) * Uu + u)) * NN;
    // Stage A: Am[m][j] = c[b,l,u][m0+m][q0c+j] (contiguous -> b128 store)
    {
      const int m = t >> 2;                      // 0..63
      const int q4 = (t & 3) * 4;                // 0..12
      float4 v = *(const float4*)&cb[(size_t)(m0 + m) * Nn + q0c + q4];
      *(float4*)&Am[m][q4] = v;
    }
    // Stage B: Bn[n][j] = cheb2[l][q0c+j][n0+n] (contiguous along n)
    {
      const int j = t >> 4;                      // 0..15
      const int n4 = (t & 15) * 4;               // 0..60
      float4 v = *(const float4*)&cheb2[(size_t)l * NN +
                                        (size_t)(q0c + j) * Nn + n0 + n4];
      Bn[n4 + 0][j] = v.x; Bn[n4 + 1][j] = v.y;
      Bn[n4 + 2][j] = v.z; Bn[n4 + 3][j] = v.w;
    }
    __syncthreads();
#pragma unroll
    for (int kk = 0; kk < 4; ++kk) {
      const int p = kk * 4 + hi * 2;
      v2f av = *(const v2f*)&Am[row][p];
      v2f b0 = *(const v2f*)&Bn[c0][p];
      v2f b1 = *(const v2f*)&Bn[c0 + 16][p];
      acc0 = wmma4(av, b0, acc0);
      acc1 = wmma4(av, b1, acc1);
    }
    __syncthreads();
  }

  float* op = out + (size_t)bu * NN;
#pragma unroll
  for (int r = 0; r < 8; ++r) {
    const int mg = m0 + tm * 16 + r + 8 * hi;
    op[(size_t)mg * Nn + n0 + c0] = acc0[r];
    op[(size_t)mg * Nn + n0 + c0 + 16] = acc1[r];
  }
}

// ---------------------------------------------------------------------------
extern "C" void kernel_launch(void* const* d_in, const int* in_sizes, int n_in,
                              void* d_out, int out_size, void* d_ws,
                              size_t ws_size, hipStream_t stream) {
  const float* x     = (const float*)d_in[0];  // [B, C, N1, N2]
  const float* cheb1 = (const float*)d_in[1];  // [K, N1, N1]
  const float* cheb2 = (const float*)d_in[2];  // [K, N2, N2]
  const float* coefs = (const float*)d_in[3];  // [K, K, C, U]

  float* a_ws = (float*)d_ws;                       // [B, K, C, N1, N2]
  float* c_ws = a_ws + (size_t)Bb * Kp * Cc * NN;   // [B, K, U, N1, N2]
  float* outp = (float*)d_out;                      // [B, U, N1, N2]

  phase1_kernel<<<dim3(16, Bb * Kp * Cc), 256, 0, stream>>>(cheb1, x, a_ws);
  phase2_kernel<<<dim3(NN / 128, Kp, Bb), 256, 0, stream>>>(coefs, a_ws, c_ws);
  phase3_kernel<<<dim3(16, Bb * Uu), 256, 0, stream>>>(c_ws, cheb2, outp);
}